// LinearAttention_29557964931173
// MI455X (gfx1250) — compile-verified
//
#include <hip/hip_runtime.h>
#include <hip/hip_bf16.h>

// ---------------------------------------------------------------------------
// Linear attention (elu+1 feature map), fp32, on MI455X (gfx1250).
//   Pass 1: kv[b,h] = phi(K)^T V  (64x64), z[b,h] = colsum(phi(K))   [M-reduction]
//   Pass 2: out = phi(Q) @ kv, row-normalized by phi(q).z + 1e-6
// Both GEMMs use V_WMMA_F32_16X16X4_F32 (full fp32 precision; problem is
// HBM-bound at AI~16 flop/byte, fp32 WMMA rate is sufficient to saturate).
// ---------------------------------------------------------------------------

typedef __attribute__((ext_vector_type(2))) float v2f;
typedef __attribute__((ext_vector_type(8))) float v8f;

namespace {
constexpr int B  = 4;
constexpr int N  = 4096;
constexpr int M  = 4096;
constexpr int H  = 8;
constexpr int D  = 64;
constexpr int DV = 64;

constexpr int NCHUNK  = 8;            // M-chunks for pass-1 parallelism
constexpr int CHUNK_M = M / NCHUNK;   // 512
constexpr int SUB_M   = 64;           // LDS staging tile (rows of k/v)
constexpr int LDSS    = D + 4;        // padded LDS row stride (68 floats)
}

__device__ __forceinline__ float elu1(float x) {
    // elu(x) + 1 :  x>0 -> x+1 ;  x<=0 -> exp(x)
    return x > 0.0f ? x + 1.0f : __expf(x);
}

// D = A(16x4) x B(4x16) + C, fp32.  A/B K-dim split across lane halves:
// vgpr0 = K{0 | 2}, vgpr1 = K{1 | 3} for lanes 0-15 | 16-31.
__device__ __forceinline__ v8f wmma_f32_4(v2f a, v2f b, v8f c) {
    return __builtin_amdgcn_wmma_f32_16x16x4_f32(
        /*neg_a=*/false, a, /*neg_b=*/false, b,
        /*c_mod=*/(short)0, c, /*reuse_a=*/false, /*reuse_b=*/false);
}

// ---------------------------------------------------------------------------
// Pass 1: partial kv (D x DV) and partial z per (chunk, b*h)
// ---------------------------------------------------------------------------
__global__ __launch_bounds__(256) void la_pass1_partial(
    const float* __restrict__ k, const float* __restrict__ v,
    const unsigned char* __restrict__ kv_mask,
    float* __restrict__ pkv, float* __restrict__ pz)
{
    const int chunk = blockIdx.x;      // 0..NCHUNK-1
    const int bh    = blockIdx.y;      // 0..B*H-1
    const int b = bh / H, h = bh % H;

    const int t    = threadIdx.x;
    const int lane = t & 31;
    const int w    = t >> 5;           // wave 0..7
    const int lr   = lane & 15;
    const int half = lane >> 4;        // K half (0: K0/K1, 1: K2/K3)
    const int dt   = w & 3;            // d-tile 0..3   (rows of kv)
    const int et0  = (w >> 2) * 2;     // e-tile pair base (0 or 2)

    __shared__ float ks[SUB_M][LDSS];  // phi(k) (masked), [m][d]
    __shared__ float vs[SUB_M][LDSS];  // v (masked),      [m][e]

    v8f c0 = {};                       // kv tile (dt, et0)
    v8f c1 = {};                       // kv tile (dt, et0+1)
    float zacc = 0.0f;

    const int r0 = t >> 4;             // staging row    0..15
    const int c4 = (t & 15) * 4;       // staging column (float4)

    for (int s = 0; s < CHUNK_M / SUB_M; ++s) {
        const int mbase = chunk * CHUNK_M + s * SUB_M;

        // ---- stage 64x64 tiles of phi(k)*mask and v*mask (coalesced b128) --
        #pragma unroll
        for (int i = 0; i < 4; ++i) {
            const int row = r0 + i * 16;
            const int m   = mbase + row;
            const float msk = kv_mask[(size_t)b * M + m] ? 1.0f : 0.0f;
            const size_t base = (((size_t)b * M + m) * H + h) * (size_t)D + c4;
            const float4 k4 = *(const float4*)(k + base);
            const float4 v4 = *(const float4*)(v + base);
            ks[row][c4 + 0] = elu1(k4.x) * msk;
            ks[row][c4 + 1] = elu1(k4.y) * msk;
            ks[row][c4 + 2] = elu1(k4.z) * msk;
            ks[row][c4 + 3] = elu1(k4.w) * msk;
            vs[row][c4 + 0] = v4.x * msk;
            vs[row][c4 + 1] = v4.y * msk;
            vs[row][c4 + 2] = v4.z * msk;
            vs[row][c4 + 3] = v4.w * msk;
        }
        __syncthreads();

        // ---- z partial: thread d sums its column of phi(k) ----------------
        if (t < D) {
            float sz = 0.0f;
            #pragma unroll 8
            for (int m = 0; m < SUB_M; ++m) sz += ks[m][t];
            zacc += sz;
        }

        // ---- WMMA accumulation: kv += phi(k)^T v over 16 K-steps of 4 -----
        #pragma unroll 4
        for (int m0 = 0; m0 < SUB_M; m0 += 4) {
            const int ma = m0 + 2 * half;
            v2f a, b0, b1;
            // A = phi(k)^T : row d = dt*16+lr, K = m
            a.x  = ks[ma    ][dt * 16 + lr];
            a.y  = ks[ma + 1][dt * 16 + lr];
            // B = v : row K = m, col e
            b0.x = vs[ma    ][et0 * 16 + lr];
            b0.y = vs[ma + 1][et0 * 16 + lr];
            b1.x = vs[ma    ][(et0 + 1) * 16 + lr];
            b1.y = vs[ma + 1][(et0 + 1) * 16 + lr];
            c0 = wmma_f32_4(a, b0, c0);
            c1 = wmma_f32_4(a, b1, c1);
        }
        __syncthreads();
    }

    // ---- store partials: C layout -> d = dt*16 + r + 8*half, e = et*16+lr -
    float* outp = pkv + ((size_t)chunk * (B * H) + bh) * (size_t)(D * DV);
    #pragma unroll
    for (int r = 0; r < 8; ++r) {
        const int d = dt * 16 + r + 8 * half;
        outp[(size_t)d * DV + et0 * 16 + lr]       = c0[r];
        outp[(size_t)d * DV + (et0 + 1) * 16 + lr] = c1[r];
    }
    if (t < D) pz[((size_t)chunk * (B * H) + bh) * D + t] = zacc;
}

// ---------------------------------------------------------------------------
// Reduce partials across chunks
// ---------------------------------------------------------------------------
__global__ __launch_bounds__(256) void la_reduce(
    const float* __restrict__ pkv, const float* __restrict__ pz,
    float* __restrict__ kvf, float* __restrict__ zf)
{
    const int bh = blockIdx.x;
    const int t  = threadIdx.x;
    #pragma unroll
    for (int j = 0; j < (D * DV) / 256; ++j) {
        const int idx = t + j * 256;
        float s = 0.0f;
        #pragma unroll
        for (int c = 0; c < NCHUNK; ++c)
            s += pkv[((size_t)c * (B * H) + bh) * (size_t)(D * DV) + idx];
        kvf[(size_t)bh * (D * DV) + idx] = s;
    }
    if (t < D) {
        float s = 0.0f;
        #pragma unroll
        for (int c = 0; c < NCHUNK; ++c)
            s += pz[((size_t)c * (B * H) + bh) * D + t];
        zf[(size_t)bh * D + t] = s;
    }
}

// ---------------------------------------------------------------------------
// Pass 2: out = phi(Q) @ kv, normalized by phi(q).z + 1e-6
// ---------------------------------------------------------------------------
__global__ __launch_bounds__(256) void la_pass2(
    const float* __restrict__ q, const float* __restrict__ kvf,
    const float* __restrict__ zf, float* __restrict__ outg)
{
    const int nt = blockIdx.x;         // 64-row n tile: 0..N/64-1
    const int bh = blockIdx.y;
    const int b = bh / H, h = bh % H;

    const int t    = threadIdx.x;
    const int lane = t & 31;
    const int w    = t >> 5;
    const int lr   = lane & 15;
    const int half = lane >> 4;
    const int nl   = w & 3;            // n sub-tile within block
    const int et0  = (w >> 2) * 2;

    __shared__ float qs[64][LDSS];     // phi(q) [n][d]
    __shared__ float kvs[D * DV];      // kv flat [d][e]
    __shared__ float zs[D];
    __shared__ float norms[64];

    const int n0 = nt * 64;
    const int r0 = t >> 4;
    const int c4 = (t & 15) * 4;

    // ---- stage phi(q) tile -------------------------------------------------
    #pragma unroll
    for (int i = 0; i < 4; ++i) {
        const int row = r0 + i * 16;
        const size_t base = (((size_t)b * N + (n0 + row)) * H + h) * (size_t)D + c4;
        const float4 q4 = *(const float4*)(q + base);
        qs[row][c4 + 0] = elu1(q4.x);
        qs[row][c4 + 1] = elu1(q4.y);
        qs[row][c4 + 2] = elu1(q4.z);
        qs[row][c4 + 3] = elu1(q4.w);
    }
    // ---- stage kv + z ------------------------------------------------------
    {
        const float4 kv4 = *(const float4*)(kvf + (size_t)bh * (D * DV) + t * 4);
        kvs[t * 4 + 0] = kv4.x; kvs[t * 4 + 1] = kv4.y;
        kvs[t * 4 + 2] = kv4.z; kvs[t * 4 + 3] = kv4.w;
    }
    if (t < D) zs[t] = zf[(size_t)bh * D + t];
    __syncthreads();

    // ---- per-row normalizer ------------------------------------------------
    if (t < 64) {
        float dot = 0.0f;
        #pragma unroll 8
        for (int d = 0; d < D; ++d) dot += qs[t][d] * zs[d];
        norms[t] = 1.0f / (dot + 1e-6f);
    }
    __syncthreads();

    // ---- GEMM: 16 K-steps of 4 over d -------------------------------------
    v8f c0 = {}, c1 = {};
    const int nrow = nl * 16 + lr;     // A row (n) for this lane
    #pragma unroll 4
    for (int d0 = 0; d0 < D; d0 += 4) {
        const int da = d0 + 2 * half;
        const v2f a = *(const v2f*)&qs[nrow][da];   // contiguous (d, d+1)
        v2f b0, b1;
        b0.x = kvs[(da    ) * DV + et0 * 16 + lr];
        b0.y = kvs[(da + 1) * DV + et0 * 16 + lr];
        b1.x = kvs[(da    ) * DV + (et0 + 1) * 16 + lr];
        b1.y = kvs[(da + 1) * DV + (et0 + 1) * 16 + lr];
        c0 = wmma_f32_4(a, b0, c0);
        c1 = wmma_f32_4(a, b1, c1);
    }

    // ---- scaled coalesced stores ------------------------------------------
    #pragma unroll
    for (int r = 0; r < 8; ++r) {
        const int nr = nl * 16 + r + 8 * half;     // C layout row
        const int n  = n0 + nr;
        const float sc = norms[nr];
        const size_t base = (((size_t)b * N + n) * H + h) * (size_t)DV;
        outg[base + et0 * 16 + lr]       = c0[r] * sc;
        outg[base + (et0 + 1) * 16 + lr] = c1[r] * sc;
    }
}

// ---------------------------------------------------------------------------
extern "C" void kernel_launch(void* const* d_in, const int* in_sizes, int n_in,
                              void* d_out, int out_size, void* d_ws, size_t ws_size,
                              hipStream_t stream) {
    const float* q = (const float*)d_in[0];
    const float* k = (const float*)d_in[1];
    const float* v = (const float*)d_in[2];
    // d_in[3] = q_mask (unused by the reference math)
    const unsigned char* kv_mask = (const unsigned char*)d_in[4];
    float* out = (float*)d_out;

    float* ws  = (float*)d_ws;
    float* pkv = ws;                                           // NCHUNK*32*4096
    float* pz  = pkv + (size_t)NCHUNK * (B * H) * (D * DV);    // NCHUNK*32*64
    float* kvf = pz  + (size_t)NCHUNK * (B * H) * D;           // 32*4096
    float* zf  = kvf + (size_t)(B * H) * (D * DV);             // 32*64

    la_pass1_partial<<<dim3(NCHUNK, B * H), 256, 0, stream>>>(k, v, kv_mask, pkv, pz);
    la_reduce<<<dim3(B * H), 256, 0, stream>>>(pkv, pz, kvf, zf);
    la_pass2<<<dim3(N / 64, B * H), 256, 0, stream>>>(q, kvf, zf, out);
}